// EK_Action_NounLoss_85074712199849
// MI455X (gfx1250) — compile-verified
//
#include <hip/hip_runtime.h>
#include <hip/hip_bf16.h>
#include <math.h>

#define NUM_ACTION 125
#define NUM_NOUN   352
#define NUM_CLS    (NUM_ACTION + NUM_NOUN)          // 477
#define BS         64
#define HW         1024                              // 32*32
#define DD         5
#define NCH        ((64 + NUM_ACTION + NUM_NOUN) * DD)  // 2705
#define BATCH_STRIDE ((size_t)NCH * HW)              // 2,769,920 floats / batch

typedef __attribute__((ext_vector_type(2))) float v2f;
typedef __attribute__((ext_vector_type(8))) float v8f;

// ---- block reductions: wave32 shuffle + 8-slot LDS stage --------------------
__device__ __forceinline__ float blk_max(float v, volatile float* sm) {
#pragma unroll
  for (int off = 16; off; off >>= 1) v = fmaxf(v, __shfl_xor(v, off, 32));
  __syncthreads();                      // protect sm from previous use
  if ((threadIdx.x & 31) == 0) sm[threadIdx.x >> 5] = v;
  __syncthreads();
  return fmaxf(fmaxf(fmaxf(sm[0], sm[1]), fmaxf(sm[2], sm[3])),
               fmaxf(fmaxf(sm[4], sm[5]), fmaxf(sm[6], sm[7])));
}

__device__ __forceinline__ float blk_sum(float v, volatile float* sm) {
#pragma unroll
  for (int off = 16; off; off >>= 1) v += __shfl_xor(v, off, 32);
  __syncthreads();
  if ((threadIdx.x & 31) == 0) sm[threadIdx.x >> 5] = v;
  __syncthreads();
  return ((sm[0] + sm[1]) + (sm[2] + sm[3])) + ((sm[4] + sm[5]) + (sm[6] + sm[7]));
}

// ---- kernel 1: per-batch argmax -> gather -> two CE losses ------------------
__global__ void __launch_bounds__(256)
EK_loss_kernel(const float* __restrict__ pred,
               const int*   __restrict__ action_gt,
               const int*   __restrict__ noun_gt,
               float*       __restrict__ ws_loss) {
  __shared__ float s_logits[NUM_CLS];
  __shared__ float s_red[8];
  __shared__ int   s_ired[8];
  __shared__ int   s_top;

  const int b = blockIdx.x;
  const int t = threadIdx.x;
  const float* bptr = pred + (size_t)b * BATCH_STRIDE;

  // Phase A: argmax of confidence (sigmoid is monotonic -> argmax raw logits).
  // s = (h*32+w)*5 + d ; value lives at channel (63*5+d), pixel hw.
  const float* cbase = bptr + (size_t)(63 * DD) * HW;
  float bv = -INFINITY;
  int   bsi = 0x7fffffff;
#pragma unroll
  for (int d = 0; d < DD; ++d) {
    for (int hw = t; hw < HW; hw += 256) {            // coalesced per d-plane
      float v = cbase[d * HW + hw];
      int   s = hw * DD + d;
      if (v > bv || (v == bv && s < bsi)) { bv = v; bsi = s; }
    }
  }
#pragma unroll
  for (int off = 16; off; off >>= 1) {
    float ov = __shfl_xor(bv, off, 32);
    int   os = __shfl_xor(bsi, off, 32);
    if (ov > bv || (ov == bv && os < bsi)) { bv = ov; bsi = os; }
  }
  if ((t & 31) == 0) { s_red[t >> 5] = bv; s_ired[t >> 5] = bsi; }
  __syncthreads();
  if (t == 0) {
    float mv = s_red[0]; int mi = s_ired[0];
#pragma unroll
    for (int w = 1; w < 8; ++w)
      if (s_red[w] > mv || (s_red[w] == mv && s_ired[w] < mi)) {
        mv = s_red[w]; mi = s_ired[w];
      }
    s_top = mi;
  }
  __syncthreads();
  const int stop   = s_top;
  const int dstar  = stop % DD;
  const int hwstar = stop / DD;

  // Phase B: gather the 477 chosen logits.
  // class c (0..476) -> channel (64+c)*5 + d*  -> offset (320 + 5c)*HW + d*HW + hw*
  const float* gbase = bptr + (size_t)(64 * DD) * HW + (size_t)dstar * HW + hwstar;
  for (int c = t; c < NUM_CLS; c += 256)
    s_logits[c] = gbase[(size_t)c * (DD * HW)];
  __syncthreads();

  // Action CE (125 classes)
  float lv = (t < NUM_ACTION) ? s_logits[t] : -INFINITY;
  float ma = blk_max(lv, s_red);
  float ea = (t < NUM_ACTION) ? expf(lv - ma) : 0.0f;
  float sa = blk_sum(ea, s_red);
  if (t == 0) {
    int lab = action_gt[b];
    ws_loss[b] = (ma + logf(sa)) - s_logits[lab];     // positive CE term
  }

  // Noun CE (352 classes): indices 125..476; all 256 threads hold one, 96 hold two
  float a0 = s_logits[NUM_ACTION + t];
  float a1 = (t < NUM_NOUN - 256) ? s_logits[NUM_ACTION + 256 + t] : -INFINITY;
  float mn = blk_max(fmaxf(a0, a1), s_red);
  float en = expf(a0 - mn) + ((t < NUM_NOUN - 256) ? expf(a1 - mn) : 0.0f);
  float sn = blk_sum(en, s_red);
  if (t == 0) {
    int lab = noun_gt[b];
    ws_loss[BS + b] = (mn + logf(sn)) - s_logits[NUM_ACTION + lab];
  }
}

// ---- kernel 2: fp32 WMMA 64-way sums + final scalars ------------------------
// A (16x4 f32) holds the 64 per-batch losses; B = ones(4x16) -> D[m,n] = row-sum.
// A layout (wave32): lane L<16 holds A[L][0..1], lane L>=16 holds A[L-16][2..3].
__global__ void __launch_bounds__(32)
EK_finish_kernel(const float* __restrict__ ws_loss, float* __restrict__ out) {
  const int L   = threadIdx.x;                 // exactly one wave32, EXEC all-1s
  const int m   = L & 15;
  const int idx = m * 4 + ((L >> 4) << 1);     // A[m][k] = v[4m + k]
  v2f aa, an, bones;
  aa[0] = ws_loss[idx];          aa[1] = ws_loss[idx + 1];
  an[0] = ws_loss[BS + idx];     an[1] = ws_loss[BS + idx + 1];
  bones[0] = 1.0f;               bones[1] = 1.0f;
  v8f cz = {0.f, 0.f, 0.f, 0.f, 0.f, 0.f, 0.f, 0.f};

  v8f da = __builtin_amdgcn_wmma_f32_16x16x4_f32(false, aa, false, bones,
                                                 (short)0, cz, false, false);
  v8f dn = __builtin_amdgcn_wmma_f32_16x16x4_f32(false, an, false, bones,
                                                 (short)0, cz, false, false);
  // lanes 0..15: rows 0..7 in da[0..7]; lanes 16..31: rows 8..15
  float pa = 0.f, pn = 0.f;
#pragma unroll
  for (int i = 0; i < 8; ++i) { pa += da[i]; pn += dn[i]; }
  pa += __shfl_xor(pa, 16, 32);
  pn += __shfl_xor(pn, 16, 32);
  if (L == 0) {
    float la = 0.5f * pa, ln = 0.5f * pn;
    out[0] = la + ln;   // total_loss
    out[1] = la;        // loss_action
    out[2] = ln;        // loss_noun
  }
}

extern "C" void kernel_launch(void* const* d_in, const int* in_sizes, int n_in,
                              void* d_out, int out_size, void* d_ws, size_t ws_size,
                              hipStream_t stream) {
  (void)in_sizes; (void)n_in; (void)out_size; (void)ws_size;
  const float* pred = (const float*)d_in[0];
  const int*   agt  = (const int*)d_in[1];
  const int*   ngt  = (const int*)d_in[2];
  float* ws  = (float*)d_ws;    // 128 floats: [0..63]=action CE, [64..127]=noun CE
  float* out = (float*)d_out;   // 3 floats

  EK_loss_kernel<<<BS, 256, 0, stream>>>(pred, agt, ngt, ws);
  EK_finish_kernel<<<1, 32, 0, stream>>>(ws, out);
}